// POD_274877907235
// MI455X (gfx1250) — compile-verified
//
#include <hip/hip_runtime.h>

typedef __attribute__((ext_vector_type(2))) float v2f;
typedef __attribute__((ext_vector_type(8))) float v8f;

static __device__ __forceinline__ float bil_zero(const float* __restrict__ p,
                                                 int H, int W, float py, float px) {
  float yf = floorf(py), xf = floorf(px);
  int y0 = (int)yf, x0 = (int)xf;
  float wy = py - yf, wx = px - xf;
  float v00 = 0.f, v01 = 0.f, v10 = 0.f, v11 = 0.f;
  bool y0ok = (y0 >= 0) && (y0 < H);
  bool y1ok = (y0 + 1 >= 0) && (y0 + 1 < H);
  bool x0ok = (x0 >= 0) && (x0 < W);
  bool x1ok = (x0 + 1 >= 0) && (x0 + 1 < W);
  if (y0ok && x0ok) v00 = p[y0 * W + x0];
  if (y0ok && x1ok) v01 = p[y0 * W + x0 + 1];
  if (y1ok && x0ok) v10 = p[(y0 + 1) * W + x0];
  if (y1ok && x1ok) v11 = p[(y0 + 1) * W + x0 + 1];
  return (v00 * (1.f - wx) + v01 * wx) * (1.f - wy) +
         (v10 * (1.f - wx) + v11 * wx) * wy;
}

// -------------------------------------------------------------------------
// Implicit-GEMM conv2d on the fp32 WMMA pipe (V_WMMA_F32_16X16X4_F32),
// LDS-staged: the block's 64-pixel B-panel is gathered into LDS once per
// (tap, ic-chunk) and consumed by all waves via ds_load_b64. K-dim order is
// (spatial tap kk OUTER, input channel ic INNER) so bounds checks / address
// math hoist out of the hot loop. Each wave owns a 16(oc) x 64(pixel) tile.
// act: 0=none, 1=relu, 2=leaky-relu(0.1)
// -------------------------------------------------------------------------
#define CONV_CH 64          // ic chunk staged per barrier
#define CONV_LDSTRIDE 68    // floats per pixel row (8B aligned, bank-friendly)

template <int KH, int KW>
__global__ __launch_bounds__(256) void conv_wmma(
    const float* __restrict__ in, const float* __restrict__ wgt,
    const float* __restrict__ bias, const float* __restrict__ res,
    float* __restrict__ out,
    int Bn, int IC, int IH, int IW, int OC, int OH, int OW,
    int stride, int pad, int act) {
  constexpr int KHW = KH * KW;
  __shared__ float smem[64 * CONV_LDSTRIDE];
  __shared__ const float* s_base[64];
  __shared__ int s_iy0[64];
  __shared__ int s_ix0[64];
  __shared__ int s_ok[64];

  const int tid = threadIdx.x;
  const int lane = tid & 31;
  const int wave = tid >> 5;
  const int nthr = blockDim.x;
  const int nwav = nthr >> 5;
  const int ocTile = blockIdx.y * nwav + wave;
  const int ocBase = ocTile * 16;
  const int npix = Bn * OH * OW;
  const int pixBase = blockIdx.x * 64;

  const int half = lane >> 4;              // 0: lanes 0-15, 1: lanes 16-31
  const int l16 = lane & 15;
  const int HWo = OH * OW;
  const size_t planeI = (size_t)IH * IW;

  // per-thread decode of its 4 output columns (needed in the epilogue)
  int bI[4], orem[4];
  bool pOK[4];
#pragma unroll
  for (int t = 0; t < 4; ++t) {
    int pix = pixBase + 16 * t + l16;
    pOK[t] = pix < npix;
    int pp = pOK[t] ? pix : 0;
    int b = pp / HWo;
    bI[t] = b;
    orem[t] = pp - b * HWo;
  }
  // first 64 threads publish the tile's pixel descriptors to LDS
  if (tid < 64) {
    int pix = pixBase + tid;
    bool ok = pix < npix;
    int pp = ok ? pix : 0;
    int b = pp / HWo;
    int r = pp - b * HWo;
    int oy = r / OW;
    int ox = r - oy * OW;
    s_iy0[tid] = oy * stride - pad;
    s_ix0[tid] = ox * stride - pad;
    s_ok[tid] = ok ? 1 : 0;
    s_base[tid] = in + (size_t)b * IC * planeI;
  }
  __syncthreads();

  // A fragment row: oc = ocBase + l16. Invalid rows read weight row 0
  // (valid memory); their accumulators are never stored.
  const int ocA = ocBase + l16;
  const bool ocOK = ocA < OC;
  const float* wrow = wgt + (size_t)(ocOK ? ocA : 0) * (size_t)IC * KHW;
  const int icSel = half ? 2 : 0;          // ISA f32 16x4 A VGPR0/1 K mapping

  v8f acc0 = {}, acc1 = {}, acc2 = {}, acc3 = {};

  for (int kk = 0; kk < KHW; ++kk) {
    const int ky = kk / KW, kx = kk - (kk / KW) * KW;
    const float* wp = wrow + kk;

    for (int c0 = 0; c0 < IC; c0 += CONV_CH) {
      const int ch = min(CONV_CH, (IC - c0 + 3) & ~3);
      // ---- cooperative stage: smem[p][i] = Im[pix p, ic c0+i] at tap kk
      for (int e = tid; e < 64 * ch; e += nthr) {
        int p = e & 63;
        int i = e >> 6;
        int gi = c0 + i;
        int iy = s_iy0[p] + ky, ix = s_ix0[p] + kx;
        bool ok = s_ok[p] && gi < IC &&
                  iy >= 0 && iy < IH && ix >= 0 && ix < IW;
        smem[p * CONV_LDSTRIDE + i] =
            ok ? s_base[p][(size_t)gi * planeI + iy * IW + ix] : 0.f;
      }
      __syncthreads();

      // ---- consume: 2 guarded weight loads + 4 ds_load_b64 + 4 WMMAs
      for (int ic = 0; ic < ch; ic += 4) {
        const int icA = ic + icSel;
        const int gA = c0 + icA;
        v2f a;
        a[0] = (gA < IC) ? wp[(size_t)gA * KHW] : 0.f;
        a[1] = (gA + 1 < IC) ? wp[(size_t)(gA + 1) * KHW] : 0.f;
        v2f b0 = *(const v2f*)&smem[(l16) * CONV_LDSTRIDE + icA];
        v2f b1 = *(const v2f*)&smem[(16 + l16) * CONV_LDSTRIDE + icA];
        v2f b2 = *(const v2f*)&smem[(32 + l16) * CONV_LDSTRIDE + icA];
        v2f b3 = *(const v2f*)&smem[(48 + l16) * CONV_LDSTRIDE + icA];
        acc0 = __builtin_amdgcn_wmma_f32_16x16x4_f32(false, a, false, b0,
                                                     (short)0, acc0, false, false);
        acc1 = __builtin_amdgcn_wmma_f32_16x16x4_f32(false, a, false, b1,
                                                     (short)0, acc1, false, false);
        acc2 = __builtin_amdgcn_wmma_f32_16x16x4_f32(false, a, false, b2,
                                                     (short)0, acc2, false, false);
        acc3 = __builtin_amdgcn_wmma_f32_16x16x4_f32(false, a, false, b3,
                                                     (short)0, acc3, false, false);
      }
      __syncthreads();
    }
  }

  // C/D layout: VGPR v -> row v (lanes 0-15) or v+8 (lanes 16-31), col = l16
#pragma unroll
  for (int v = 0; v < 8; ++v) {
    int oc = ocBase + v + (half ? 8 : 0);
    if (oc < OC) {
      float bsv = bias[oc];
#pragma unroll
      for (int t = 0; t < 4; ++t) {
        if (pOK[t]) {
          float x = (t == 0 ? acc0[v] : t == 1 ? acc1[v] : t == 2 ? acc2[v]
                                                         : acc3[v]) + bsv;
          size_t oidx = ((size_t)bI[t] * OC + oc) * (size_t)HWo + orem[t];
          if (res) x += res[oidx];
          if (act == 1) x = fmaxf(x, 0.f);
          else if (act == 2) x = x > 0.f ? x : 0.1f * x;
          out[oidx] = x;
        }
      }
    }
  }
}

// ------------------------- elementwise kernels ---------------------------

__global__ void k_zero(float* p, int n) {
  int i = blockIdx.x * blockDim.x + threadIdx.x;
  if (i < n) p[i] = 0.f;
}

__global__ void k_norm_inplace(float* p, int n) {
  int i = blockIdx.x * blockDim.x + threadIdx.x;
  if (i < n) p[i] = (p[i] - 0.45f) / 0.226f;
}

__global__ void k_avgpool2(const float* __restrict__ in, float* __restrict__ out,
                           int Bn, int C, int OH, int OW) {
  int i = blockIdx.x * blockDim.x + threadIdx.x;
  int total = Bn * C * OH * OW;
  if (i >= total) return;
  int ox = i % OW;
  int t = i / OW;
  int oy = t % OH;
  int bc = t / OH;
  int IH = OH * 2, IW = OW * 2;
  const float* p = in + (size_t)bc * IH * IW;
  int y = oy * 2, x = ox * 2;
  out[i] = 0.25f * (p[y * IW + x] + p[y * IW + x + 1] +
                    p[(y + 1) * IW + x] + p[(y + 1) * IW + x + 1]);
}

// bilinear resize with border clamp; alignC selects align_corners mode;
// postScale fuses the "*2" applied to upsampled flows / features.
__global__ void k_resize(const float* __restrict__ in, float* __restrict__ out,
                         int Bn, int C, int IH, int IW, int OH, int OW,
                         int alignC, float postScale) {
  int i = blockIdx.x * blockDim.x + threadIdx.x;
  int total = Bn * C * OH * OW;
  if (i >= total) return;
  int ox = i % OW;
  int t = i / OW;
  int oy = t % OH;
  int bc = t / OH;
  float sy, sx;
  if (alignC) {
    sy = (OH > 1) ? oy * ((float)(IH - 1) / (float)(OH - 1)) : 0.f;
    sx = (OW > 1) ? ox * ((float)(IW - 1) / (float)(OW - 1)) : 0.f;
  } else {
    sy = fminf(fmaxf((oy + 0.5f) * ((float)IH / (float)OH) - 0.5f, 0.f),
               (float)(IH - 1));
    sx = fminf(fmaxf((ox + 0.5f) * ((float)IW / (float)OW) - 0.5f, 0.f),
               (float)(IW - 1));
  }
  int y0 = (int)floorf(sy), x0 = (int)floorf(sx);
  int y1 = min(y0 + 1, IH - 1), x1 = min(x0 + 1, IW - 1);
  float wy = sy - (float)y0, wx = sx - (float)x0;
  const float* p = in + (size_t)bc * IH * IW;
  float v00 = p[y0 * IW + x0], v01 = p[y0 * IW + x1];
  float v10 = p[y1 * IW + x0], v11 = p[y1 * IW + x1];
  float r0 = v00 * (1.f - wx) + v01 * wx;
  float r1 = v10 * (1.f - wx) + v11 * wx;
  out[i] = (r0 * (1.f - wy) + r1 * wy) * postScale;
}

// zero-padding bilinear warp: out[b,c,y,x] = x[b,c, y+flow_y, x+flow_x]
__global__ void k_flowwarp(const float* __restrict__ x,
                           const float* __restrict__ flow,
                           float* __restrict__ out, int Bn, int C, int H, int W) {
  int i = blockIdx.x * blockDim.x + threadIdx.x;
  int total = Bn * C * H * W;
  if (i >= total) return;
  int p = i % (H * W);
  int t = i / (H * W);
  int c = t % C;
  int b = t / C;
  int y = p / W, xx = p % W;
  float fx = flow[((size_t)b * 2 + 0) * H * W + p];
  float fy = flow[((size_t)b * 2 + 1) * H * W + p];
  out[i] = bil_zero(x + ((size_t)b * C + c) * H * W, H, W,
                    (float)y + fy, (float)xx + fx);
}

// channel-slice copy for concatenation
__global__ void k_copych(const float* __restrict__ src, float* __restrict__ dst,
                         int Bn, int Cs, int HW, int Cd, int co) {
  int i = blockIdx.x * blockDim.x + threadIdx.x;
  int total = Bn * Cs * HW;
  if (i >= total) return;
  int p = i % HW;
  int t = i / HW;
  int c = t % Cs;
  int b = t / Cs;
  dst[((size_t)b * Cd + co + c) * HW + p] = src[i];
}

// Deformable-conv sampling: val[b, c*9+k2, y, x] =
//   sigmoid(mask) * bilinear_zero(x[b,c], y + ky + dy, x + kx + dx)
// raw = offset-conv output [B, 3*DG*9, H, W] (dy | dx | mask)
__global__ void k_deform_sample(const float* __restrict__ x,
                                const float* __restrict__ raw,
                                float* __restrict__ val,
                                int Bn, int C, int Hs, int Ws, int DGc, int Cg) {
  int i = blockIdx.x * blockDim.x + threadIdx.x;
  int total = Bn * C * 9 * Hs * Ws;
  if (i >= total) return;
  int HW = Hs * Ws;
  int p = i % HW;
  int t = i / HW;
  int k2 = t % 9;
  t /= 9;
  int c = t % C;
  int b = t / C;
  int y = p / Ws, xx = p % Ws;
  int dg = c / Cg;
  int n72 = DGc * 9;
  size_t base = (((size_t)b * 3 * n72) + dg * 9 + k2) * HW + p;
  float dy = raw[base];
  float dx = raw[base + (size_t)n72 * HW];
  float mk = raw[base + (size_t)2 * n72 * HW];
  float m = 1.f / (1.f + expf(-mk));
  int ky = k2 / 3 - 1, kx = k2 % 3 - 1;
  float v = bil_zero(x + ((size_t)b * C + c) * HW, Hs, Ws,
                     (float)(y + ky) + dy, (float)(xx + kx) + dx);
  val[i] = v * m;
}

// ------------------------------ host side --------------------------------

static inline int cdiv(int a, int b) { return (a + b - 1) / b; }

extern "C" void kernel_launch(void* const* d_in, const int* in_sizes, int n_in,
                              void* d_out, int out_size, void* d_ws, size_t ws_size,
                              hipStream_t stream) {
  (void)in_sizes; (void)n_in; (void)out_size; (void)ws_size;

  const int B_ = 2, Hh = 128, Ww = 128, NF_ = 64, DG_ = 8, CG_ = NF_ / DG_;

  const float* nbr1 = (const float*)d_in[0];
  const float* nbr2 = (const float*)d_in[1];
  const float* nbr3 = (const float*)d_in[2];
  const float* ref1 = (const float*)d_in[3];
  const float* ref2 = (const float*)d_in[4];
  const float* ref3 = (const float*)d_in[5];
  const float* nfr  = (const float*)d_in[6];
  const float* rfr  = (const float*)d_in[7];

  // params dict flattened (w,b) pairs in insertion order:
  enum {
    P_fds0_1 = 0, P_fds0_2, P_fds1_1, P_fds1_2, P_fds2_1, P_fds2_2,
    P_L3_oc1, P_L3_dcn_off, P_L3_dcn,
    P_L2_oc1, P_L2_dcn_off, P_L2_dcn,
    P_L1_oc1, P_L1_dcn_off, P_L1_dcn,
    P_L3_oc2,
    P_L2_oc2, P_L2_oc3, P_L2_fc,
    P_L1_oc2, P_L1_oc3, P_L1_fc,
    P_cas_oc1, P_cas_oc2, P_cas_dcn_off, P_cas_dcn,
    P_COUNT
  };
  auto PW = [&](int i) { return (const float*)d_in[8 + 2 * i]; };
  auto PB = [&](int i) { return (const float*)d_in[8 + 2 * i + 1]; };
  const int SPY_BASE = 8 + 2 * P_COUNT;  // 60
  auto SW = [&](int l, int i) { return (const float*)d_in[SPY_BASE + 2 * (l * 5 + i)]; };
  auto SB = [&](int l, int i) { return (const float*)d_in[SPY_BASE + 2 * (l * 5 + i) + 1]; };

  // ---- workspace bump allocator (deterministic every call) ----
  char* wsb = (char*)d_ws;
  size_t cur = 0;
  auto take = [&](size_t nf) -> float* {
    float* p = (float*)(wsb + cur);
    cur += ((nf * sizeof(float) + 255) / 256) * 256;
    return p;
  };

  auto ew = [&](int n) { return dim3(cdiv(n, 256)); };

  // generic conv launcher with template dispatch on kernel size
  auto convG = [&](const float* in, int IC, int IH, int IW,
                   const float* w, const float* b, float* out, int OC,
                   int KH, int stride, int pad, int act,
                   const float* res = nullptr) {
    int OH = (IH + 2 * pad - KH) / stride + 1;
    int OW = (IW + 2 * pad - KH) / stride + 1;
    int npix = B_ * OH * OW;
    int tilesOC = cdiv(OC, 16);
    int W = tilesOC < 8 ? tilesOC : 8;
    dim3 grid(cdiv(npix, 64), cdiv(tilesOC, W));
    dim3 blk(32 * W);
    if (KH == 7)
      conv_wmma<7, 7><<<grid, blk, 0, stream>>>(in, w, b, res, out, B_, IC, IH,
                                                IW, OC, OH, OW, stride, pad, act);
    else if (KH == 3)
      conv_wmma<3, 3><<<grid, blk, 0, stream>>>(in, w, b, res, out, B_, IC, IH,
                                                IW, OC, OH, OW, stride, pad, act);
    else
      conv_wmma<1, 1><<<grid, blk, 0, stream>>>(in, w, b, res, out, B_, IC, IH,
                                                IW, OC, OH, OW, stride, pad, act);
  };
  auto conv = [&](const float* in, int IC, int IH, int IW, int prm,
                  float* out, int OC, int KH, int stride, int pad, int act,
                  const float* res = nullptr) {
    convG(in, IC, IH, IW, PW(prm), PB(prm), out, OC, KH, stride, pad, act, res);
  };
  auto convSpy = [&](const float* in, int IC, int s, int l, int i,
                     float* out, int OC, int act, const float* res = nullptr) {
    convG(in, IC, s, s, SW(l, i), SB(l, i), out, OC, 7, 1, 3, act, res);
  };

  // ---- persistent buffers ----
  float* pyr_ref[6];
  float* pyr_supp[6];
  for (int l = 0; l < 6; ++l) {
    int s = 16 << l;
    pyr_ref[l] = take((size_t)B_ * s * s);
    pyr_supp[l] = take((size_t)B_ * s * s);
  }
  float* flowA = take((size_t)B_ * 2 * 512 * 512);
  float* flowB = take((size_t)B_ * 2 * 512 * 512);
  float* upB   = take((size_t)B_ * 2 * 512 * 512);
  float* warpB = take((size_t)B_ * 1 * 512 * 512);
  // big scratch: sA >= 64ch@512^2, sB/sC >= 32ch@512^2 (reused all phases)
  float* sA = take((size_t)B_ * 64 * 512 * 512);
  float* sB = take((size_t)B_ * 32 * 512 * 512);
  float* sC = take((size_t)B_ * 32 * 512 * 512);

  float* f1 = take((size_t)B_ * 16 * 128 * 128);
  float* f2 = take((size_t)B_ * 16 * 128 * 128);
  float* l2f1 = take((size_t)B_ * 16 * 64 * 64);
  float* l2f2 = take((size_t)B_ * 16 * 64 * 64);
  float* l3f1 = take((size_t)B_ * 16 * 32 * 32);
  float* l3f2 = take((size_t)B_ * 16 * 32 * 32);
  float* nfr2 = take((size_t)B_ * 64 * 64);
  float* rfr2 = take((size_t)B_ * 64 * 64);
  float* nfr3 = take((size_t)B_ * 32 * 32);
  float* rfr3 = take((size_t)B_ * 32 * 32);
  float* l3o_a = take((size_t)B_ * NF_ * 32 * 32);
  float* l3o   = take((size_t)B_ * NF_ * 32 * 32);
  float* l3fea = take((size_t)B_ * NF_ * 32 * 32);
  float* l2o_a = take((size_t)B_ * NF_ * 64 * 64);
  float* l3o_up = take((size_t)B_ * NF_ * 64 * 64);
  float* l2o_b = take((size_t)B_ * NF_ * 64 * 64);
  float* l2o   = take((size_t)B_ * NF_ * 64 * 64);
  float* l2fea_pre = take((size_t)B_ * NF_ * 64 * 64);
  float* l3fea_up  = take((size_t)B_ * NF_ * 64 * 64);
  float* l2fea     = take((size_t)B_ * NF_ * 64 * 64);
  float* l1o_a  = take((size_t)B_ * NF_ * 128 * 128);
  float* l2o_up = take((size_t)B_ * NF_ * 128 * 128);
  float* l1o_b  = take((size_t)B_ * NF_ * 128 * 128);
  float* l1o    = take((size_t)B_ * NF_ * 128 * 128);
  float* l1fea_pre = take((size_t)B_ * NF_ * 128 * 128);
  float* l2fea_up  = take((size_t)B_ * NF_ * 128 * 128);
  float* l1fea     = take((size_t)B_ * NF_ * 128 * 128);
  float* off_a     = take((size_t)B_ * NF_ * 128 * 128);
  float* offb      = take((size_t)B_ * NF_ * 128 * 128);

  // ================= SPyNet flow =================
  k_resize<<<ew(B_ * 512 * 512), 256, 0, stream>>>(nfr, pyr_ref[5], B_, 1, Hh, Ww,
                                                   512, 512, 0, 1.f);
  k_norm_inplace<<<ew(B_ * 512 * 512), 256, 0, stream>>>(pyr_ref[5], B_ * 512 * 512);
  k_resize<<<ew(B_ * 512 * 512), 256, 0, stream>>>(rfr, pyr_supp[5], B_, 1, Hh, Ww,
                                                   512, 512, 0, 1.f);
  k_norm_inplace<<<ew(B_ * 512 * 512), 256, 0, stream>>>(pyr_supp[5], B_ * 512 * 512);
  for (int l = 5; l >= 1; --l) {
    int so = 16 << (l - 1);
    k_avgpool2<<<ew(B_ * so * so), 256, 0, stream>>>(pyr_ref[l], pyr_ref[l - 1], B_, 1, so, so);
    k_avgpool2<<<ew(B_ * so * so), 256, 0, stream>>>(pyr_supp[l], pyr_supp[l - 1], B_, 1, so, so);
  }

  float* flowCur = flowA;
  float* flowNxt = flowB;
  k_zero<<<ew(B_ * 2 * 8 * 8), 256, 0, stream>>>(flowCur, B_ * 2 * 8 * 8);
  for (int l = 0; l < 6; ++l) {
    int s = 16 << l, hs = s / 2, HWs = s * s;
    k_resize<<<ew(B_ * 2 * HWs), 256, 0, stream>>>(flowCur, upB, B_, 2, hs, hs, s, s, 1, 2.f);
    k_flowwarp<<<ew(B_ * HWs), 256, 0, stream>>>(pyr_supp[l], upB, warpB, B_, 1, s, s);
    k_copych<<<ew(B_ * 1 * HWs), 256, 0, stream>>>(pyr_ref[l], sC, B_, 1, HWs, 4, 0);
    k_copych<<<ew(B_ * 1 * HWs), 256, 0, stream>>>(warpB, sC, B_, 1, HWs, 4, 1);
    k_copych<<<ew(B_ * 2 * HWs), 256, 0, stream>>>(upB, sC, B_, 2, HWs, 4, 2);
    convSpy(sC, 4, s, l, 0, sB, 32, 1);
    convSpy(sB, 32, s, l, 1, sA, 64, 1);
    convSpy(sA, 64, s, l, 2, sC, 32, 1);
    convSpy(sC, 32, s, l, 3, sB, 16, 1);
    convSpy(sB, 16, s, l, 4, flowNxt, 2, 0, upB);
    float* t = flowCur; flowCur = flowNxt; flowNxt = t;
  }
  // flowCur: [B,2,512,512]

  // ================= flow downsampling features =================
  conv(flowCur, 2, 512, 512, P_fds0_1, f1, 16, 3, 4, 1, 0);
  conv(flowCur, 2, 512, 512, P_fds0_2, f2, 16, 3, 4, 1, 0);
  conv(f1, 16, 128, 128, P_fds1_1, l2f1, 16, 3, 2, 1, 0);
  conv(f2, 16, 128, 128, P_fds1_2, l2f2, 16, 3, 2, 1, 0);
  conv(l2f1, 16, 64, 64, P_fds2_1, l3f1, 16, 3, 2, 1, 0);
  conv(l2f2, 16, 64, 64, P_fds2_2, l3f2, 16, 3, 2, 1, 0);
  k_resize<<<ew(B_ * 64 * 64), 256, 0, stream>>>(nfr, nfr2, B_, 1, 128, 128, 64, 64, 0, 1.f);
  k_resize<<<ew(B_ * 64 * 64), 256, 0, stream>>>(rfr, rfr2, B_, 1, 128, 128, 64, 64, 0, 1.f);
  k_resize<<<ew(B_ * 32 * 32), 256, 0, stream>>>(nfr2, nfr3, B_, 1, 64, 64, 32, 32, 0, 1.f);
  k_resize<<<ew(B_ * 32 * 32), 256, 0, stream>>>(rfr2, rfr3, B_, 1, 64, 64, 32, 32, 0, 1.f);

  // ================= L3 =================
  {
    int s = 32, HWs = s * s;
    k_copych<<<ew(B_ * 64 * HWs), 256, 0, stream>>>(nbr3, sB, B_, 64, HWs, 162, 0);
    k_copych<<<ew(B_ * 64 * HWs), 256, 0, stream>>>(ref3, sB, B_, 64, HWs, 162, 64);
    k_copych<<<ew(B_ * 16 * HWs), 256, 0, stream>>>(l3f1, sB, B_, 16, HWs, 162, 128);
    k_copych<<<ew(B_ * 16 * HWs), 256, 0, stream>>>(l3f2, sB, B_, 16, HWs, 162, 144);
    k_copych<<<ew(B_ * 1 * HWs), 256, 0, stream>>>(nfr3, sB, B_, 1, HWs, 162, 160);
    k_copych<<<ew(B_ * 1 * HWs), 256, 0, stream>>>(rfr3, sB, B_, 1, HWs, 162, 161);
    conv(sB, 162, s, s, P_L3_oc1, l3o_a, NF_, 3, 1, 1, 2);
    conv(l3o_a, NF_, s, s, P_L3_oc2, l3o, NF_, 3, 1, 1, 2);
    conv(l3o, NF_, s, s, P_L3_dcn_off, sC, DG_ * 3 * 9, 3, 1, 1, 0);
    k_deform_sample<<<ew(B_ * NF_ * 9 * HWs), 256, 0, stream>>>(nbr3, sC, sA,
                                                                B_, NF_, s, s, DG_, CG_);
    conv(sA, NF_ * 9, s, s, P_L3_dcn, l3fea, NF_, 1, 1, 0, 2);  // lrelu fused
  }

  // ================= L2 =================
  {
    int s = 64, HWs = s * s;
    k_copych<<<ew(B_ * 64 * HWs), 256, 0, stream>>>(nbr2, sB, B_, 64, HWs, 162, 0);
    k_copych<<<ew(B_ * 64 * HWs), 256, 0, stream>>>(ref2, sB, B_, 64, HWs, 162, 64);
    k_copych<<<ew(B_ * 16 * HWs), 256, 0, stream>>>(l2f1, sB, B_, 16, HWs, 162, 128);
    k_copych<<<ew(B_ * 16 * HWs), 256, 0, stream>>>(l2f2, sB, B_, 16, HWs, 162, 144);
    k_copych<<<ew(B_ * 1 * HWs), 256, 0, stream>>>(nfr2, sB, B_, 1, HWs, 162, 160);
    k_copych<<<ew(B_ * 1 * HWs), 256, 0, stream>>>(rfr2, sB, B_, 1, HWs, 162, 161);
    conv(sB, 162, s, s, P_L2_oc1, l2o_a, NF_, 3, 1, 1, 2);
    k_resize<<<ew(B_ * NF_ * HWs), 256, 0, stream>>>(l3o, l3o_up, B_, NF_, 32, 32, s, s, 0, 2.f);
    k_copych<<<ew(B_ * NF_ * HWs), 256, 0, stream>>>(l2o_a, sC, B_, NF_, HWs, 128, 0);
    k_copych<<<ew(B_ * NF_ * HWs), 256, 0, stream>>>(l3o_up, sC, B_, NF_, HWs, 128, 64);
    conv(sC, 128, s, s, P_L2_oc2, l2o_b, NF_, 3, 1, 1, 2);
    conv(l2o_b, NF_, s, s, P_L2_oc3, l2o, NF_, 3, 1, 1, 2);
    conv(l2o, NF_, s, s, P_L2_dcn_off, sB, DG_ * 3 * 9, 3, 1, 1, 0);
    k_deform_sample<<<ew(B_ * NF_ * 9 * HWs), 256, 0, stream>>>(nbr2, sB, sA,
                                                                B_, NF_, s, s, DG_, CG_);
    conv(sA, NF_ * 9, s, s, P_L2_dcn, l2fea_pre, NF_, 1, 1, 0, 0);  // no act
    k_resize<<<ew(B_ * NF_ * HWs), 256, 0, stream>>>(l3fea, l3fea_up, B_, NF_, 32, 32, s, s, 0, 1.f);
    k_copych<<<ew(B_ * NF_ * HWs), 256, 0, stream>>>(l2fea_pre, sC, B_, NF_, HWs, 128, 0);
    k_copych<<<ew(B_ * NF_ * HWs), 256, 0, stream>>>(l3fea_up, sC, B_, NF_, HWs, 128, 64);
    conv(sC, 128, s, s, P_L2_fc, l2fea, NF_, 3, 1, 1, 2);
  }

  // ================= L1 =================
  {
    int s = 128, HWs = s * s;
    k_copych<<<ew(B_ * 64 * HWs), 256, 0, stream>>>(nbr1, sB, B_, 64, HWs, 162, 0);
    k_copych<<<ew(B_ * 64 * HWs), 256, 0, stream>>>(ref1, sB, B_, 64, HWs, 162, 64);
    k_copych<<<ew(B_ * 16 * HWs), 256, 0, stream>>>(f1, sB, B_, 16, HWs, 162, 128);
    k_copych<<<ew(B_ * 16 * HWs), 256, 0, stream>>>(f2, sB, B_, 16, HWs, 162, 144);
    k_copych<<<ew(B_ * 1 * HWs), 256, 0, stream>>>(nfr, sB, B_, 1, HWs, 162, 160);
    k_copych<<<ew(B_ * 1 * HWs), 256, 0, stream>>>(rfr, sB, B_, 1, HWs, 162, 161);
    conv(sB, 162, s, s, P_L1_oc1, l1o_a, NF_, 3, 1, 1, 2);
    k_resize<<<ew(B_ * NF_ * HWs), 256, 0, stream>>>(l2o, l2o_up, B_, NF_, 64, 64, s, s, 0, 2.f);
    k_copych<<<ew(B_ * NF_ * HWs), 256, 0, stream>>>(l1o_a, sC, B_, NF_, HWs, 128, 0);
    k_copych<<<ew(B_ * NF_ * HWs), 256, 0, stream>>>(l2o_up, sC, B_, NF_, HWs, 128, 64);
    conv(sC, 128, s, s, P_L1_oc2, l1o_b, NF_, 3, 1, 1, 2);
    conv(l1o_b, NF_, s, s, P_L1_oc3, l1o, NF_, 3, 1, 1, 2);
    conv(l1o, NF_, s, s, P_L1_dcn_off, sB, DG_ * 3 * 9, 3, 1, 1, 0);
    k_deform_sample<<<ew(B_ * NF_ * 9 * HWs), 256, 0, stream>>>(nbr1, sB, sA,
                                                                B_, NF_, s, s, DG_, CG_);
    conv(sA, NF_ * 9, s, s, P_L1_dcn, l1fea_pre, NF_, 1, 1, 0, 0);  // no act
    k_resize<<<ew(B_ * NF_ * HWs), 256, 0, stream>>>(l2fea, l2fea_up, B_, NF_, 64, 64, s, s, 0, 1.f);
    k_copych<<<ew(B_ * NF_ * HWs), 256, 0, stream>>>(l1fea_pre, sC, B_, NF_, HWs, 128, 0);
    k_copych<<<ew(B_ * NF_ * HWs), 256, 0, stream>>>(l2fea_up, sC, B_, NF_, HWs, 128, 64);
    conv(sC, 128, s, s, P_L1_fc, l1fea, NF_, 3, 1, 1, 0);  // no act
  }

  // ================= cascade =================
  {
    int s = 128, HWs = s * s;
    k_copych<<<ew(B_ * NF_ * HWs), 256, 0, stream>>>(l1fea, sC, B_, NF_, HWs, 128, 0);
    k_copych<<<ew(B_ * NF_ * HWs), 256, 0, stream>>>(ref1, sC, B_, NF_, HWs, 128, 64);
    conv(sC, 128, s, s, P_cas_oc1, off_a, NF_, 3, 1, 1, 2);
    conv(off_a, NF_, s, s, P_cas_oc2, offb, NF_, 3, 1, 1, 2);
    conv(offb, NF_, s, s, P_cas_dcn_off, sB, DG_ * 3 * 9, 3, 1, 1, 0);
    k_deform_sample<<<ew(B_ * NF_ * 9 * HWs), 256, 0, stream>>>(l1fea, sB, sA,
                                                                B_, NF_, s, s, DG_, CG_);
    conv(sA, NF_ * 9, s, s, P_cas_dcn, (float*)d_out, NF_, 1, 1, 0, 2);  // lrelu
  }
}